// AttentionInteractionBlock_46273977647384
// MI455X (gfx1250) — compile-verified
//
#include <hip/hip_runtime.h>
#include <math.h>

// ---------------------------------------------------------------------------
// MI455X (gfx1250) implementation of the AttentionInteractionBlock.
// wave32; WMMA f32 16x16x4 for all GEMM stages; fused edge kernel with
// LDS-staged fc GEMMs and f32 global atomics for the segment sum.
// ---------------------------------------------------------------------------

#define MULC 64
#define NATTRC 4
#define NBASISC 8

typedef float v2f __attribute__((ext_vector_type(2)));
typedef float v8f __attribute__((ext_vector_type(8)));

__device__ __forceinline__ v8f wmma_f32(v2f a, v2f b, v8f c) {
  // 8 args: (neg_a, A, neg_b, B, c_mod, C, reuse_a, reuse_b)
  return __builtin_amdgcn_wmma_f32_16x16x4_f32(
      false, a, false, b, (short)0, c, false, false);
}

__device__ __forceinline__ float softplusf(float v) {
  return (v > 20.f) ? v : log1pf(__expf(v));
}
__device__ __forceinline__ float sspf(float v) {
  return (softplusf(v) - 0.69314718055994531f) * 1.675f;
}
__device__ __forceinline__ float siluf(float v) {
  return v / (1.f + __expf(-v));
}
__device__ __forceinline__ void atomic_add_f32(float* p, float v) {
  __hip_atomic_fetch_add(p, v, __ATOMIC_RELAXED, __HIP_MEMORY_SCOPE_AGENT);
}

// ---------------------------------------------------------------------------
// Kernel 0: zero the aggregation scratch (agg0 + agg1 = N*512 floats)
// ---------------------------------------------------------------------------
__global__ void init_kernel(float* __restrict__ p, size_t n) {
  size_t i = (size_t)blockIdx.x * blockDim.x + threadIdx.x;
  if (i < n) p[i] = 0.f;
}

// ---------------------------------------------------------------------------
// Kernel 1: node-side.  Block = 128 threads = 4 waves; one 16-node tile.
//   wave 0 : h0   -> sc0 (into out[:, :64]),    h0' = h0@lin1_w0/8 (ws)
//   wave m : h1_m -> sc1_m (into out[:, 64+w*3+m]), h1_m' (ws, (N,64,3))
// sc is computed as G = A @ sc_w.reshape(64,256), then sc[.,w] =
// sum_v x[.,v]*G[., v*64+w] / 16, accumulated per-fragment so G never
// leaves registers.
// ---------------------------------------------------------------------------
__global__ __launch_bounds__(128) void node_kernel(
    const float* __restrict__ x,        // (N,4)
    const float* __restrict__ h,        // (N,256)
    const float* __restrict__ lin1_w0,  // (64,64)
    const float* __restrict__ lin1_w1,  // (64,64)
    const float* __restrict__ sc_w0,    // (64,4,64) == (64,256)
    const float* __restrict__ sc_w1,    // (64,4,64)
    float* __restrict__ out,            // (N,256)
    float* __restrict__ h0p,            // (N,64)
    float* __restrict__ h1p)            // (N,64,3)
{
  __shared__ float Alds[4][16 * 64];    // 16 KB
  __shared__ float xlds[16 * 4];

  const int row0 = blockIdx.x * 16;
  const int tid  = threadIdx.x;
  const int wave = tid >> 5;
  const int lane = tid & 31;

  // Stage the four 16x64 A tiles (h0 and the 3 components of h1).
  for (int i = tid; i < 16 * 64; i += 128) {
    const int r = i >> 6, u = i & 63;
    const float* hr = h + (size_t)(row0 + r) * 256;
    Alds[0][r * 64 + u] = hr[u];
    Alds[1][r * 64 + u] = hr[64 + u * 3 + 0];
    Alds[2][r * 64 + u] = hr[64 + u * 3 + 1];
    Alds[3][r * 64 + u] = hr[64 + u * 3 + 2];
  }
  if (tid < 64) xlds[tid] = x[(size_t)row0 * 4 + tid];
  __syncthreads();

  const int   mlo    = lane & 15;        // A-row / B-col within tile
  const int   khalf  = (lane >> 4) * 2;  // K sub-pair select
  const int   rowsel = (lane >> 4) * 8;  // C/D row offset
  const float* A     = Alds[wave];
  const float* scw   = (wave == 0) ? sc_w0 : sc_w1;
  const float* linw  = (wave == 0) ? lin1_w0 : lin1_w1;

  // ---- self connection: 4 v-slices x 4 col tiles, K = 64 ----
  v8f acc[4];
  acc[0] = {}; acc[1] = {}; acc[2] = {}; acc[3] = {};
  for (int v = 0; v < 4; ++v) {
    for (int ct = 0; ct < 4; ++ct) {
      v8f c = {};
      const int colbase = v * 64 + ct * 16;
      for (int kk = 0; kk < 16; ++kk) {
        const int k0 = kk * 4 + khalf;
        v2f a; a.x = A[mlo * 64 + k0]; a.y = A[mlo * 64 + k0 + 1];
        v2f b; b.x = scw[(size_t)k0 * 256 + colbase + mlo];
               b.y = scw[(size_t)(k0 + 1) * 256 + colbase + mlo];
        c = wmma_f32(a, b, c);
      }
#pragma unroll
      for (int r = 0; r < 8; ++r)
        acc[ct][r] += xlds[(r + rowsel) * 4 + v] * c[r];
    }
  }
  const float sc_scale = 1.0f / 16.0f;   // 1/sqrt(64*4)
  for (int ct = 0; ct < 4; ++ct) {
#pragma unroll
    for (int r = 0; r < 8; ++r) {
      const int row  = row0 + r + rowsel;
      const int wcol = ct * 16 + mlo;
      const float vo = acc[ct][r] * sc_scale;
      if (wave == 0) out[(size_t)row * 256 + wcol] = vo;
      else           out[(size_t)row * 256 + 64 + wcol * 3 + (wave - 1)] = vo;
    }
  }

  // ---- lin1: 4 col tiles, K = 64 ----
  const float lin_scale = 0.125f;        // 1/sqrt(64)
  for (int ct = 0; ct < 4; ++ct) {
    v8f c = {};
    const int colbase = ct * 16;
    for (int kk = 0; kk < 16; ++kk) {
      const int k0 = kk * 4 + khalf;
      v2f a; a.x = A[mlo * 64 + k0]; a.y = A[mlo * 64 + k0 + 1];
      v2f b; b.x = linw[(size_t)k0 * 64 + colbase + mlo];
             b.y = linw[(size_t)(k0 + 1) * 64 + colbase + mlo];
      c = wmma_f32(a, b, c);
    }
#pragma unroll
    for (int r = 0; r < 8; ++r) {
      const int row  = row0 + r + rowsel;
      const int vcol = colbase + mlo;
      const float vo = c[r] * lin_scale;
      if (wave == 0) h0p[(size_t)row * 64 + vcol] = vo;
      else           h1p[(size_t)row * 192 + vcol * 3 + (wave - 1)] = vo;
    }
  }
}

// ---------------------------------------------------------------------------
// Kernel 2: fused edge kernel.  Block = 128 threads = 4 waves; each wave
// owns one 16-edge tile.  Per tile:
//   t = ssp(ele @ fc_w1 / sqrt8)         (WMMA, K=8)
//   w = t @ fc_w2 / sqrt8                (WMMA, one u-group of cols at a time)
//   att = MLP(sin(n pi r)/r)             (lanes 0..15)
//   p00/p11/p01/p10 products with gathered h0'/h1' -> atomic segment sum
// ---------------------------------------------------------------------------
__global__ __launch_bounds__(128) void edge_kernel(
    const float* __restrict__ ele,       // (E,8)
    const float* __restrict__ edge_sh,   // (E,4)
    const float* __restrict__ r_ijs,     // (E)
    const float* __restrict__ fc_w1,     // (8,8)
    const float* __restrict__ fc_w2,     // (8,256)
    const float* __restrict__ raw1, const float* __restrict__ rab1,
    const float* __restrict__ raw2, const float* __restrict__ rab2,
    const float* __restrict__ raw3, const float* __restrict__ rab3,
    const float* __restrict__ raw4, const float* __restrict__ rab4,
    const int*   __restrict__ edge_index,// (2,E)
    const float* __restrict__ h0p,       // (N,64)
    const float* __restrict__ h1p,       // (N,64,3)
    float* __restrict__ agg0,            // (N,128)
    float* __restrict__ agg1,            // (N,128,3)
    int E)
{
  __shared__ float Aele[4][16 * 8];
  __shared__ float Tlds[4][16 * 8];
  __shared__ float attl[4][16];
  __shared__ float wbuf[4][4 * 16 * 16]; // [wave][j][e][n]  (w slices)

  const int wave = threadIdx.x >> 5;
  const int lane = threadIdx.x & 31;
  const int e0   = (blockIdx.x * 4 + wave) * 16;

  for (int i = lane; i < 16 * 8; i += 32)
    Aele[wave][i] = ele[(size_t)e0 * 8 + i];

  if (lane < 16) {                       // attention MLP, one edge per lane
    const float r = r_ijs[e0 + lane];
    float b0[8], a1[8], a2[8], a3[8];
#pragma unroll
    for (int i = 0; i < 8; ++i)
      b0[i] = __sinf((float)(i + 1) * 3.14159265358979323846f * r) / r;
#pragma unroll
    for (int j = 0; j < 8; ++j) {
      float s = rab1[j];
      for (int i = 0; i < 8; ++i) s += b0[i] * raw1[i * 8 + j];
      a1[j] = siluf(s);
    }
#pragma unroll
    for (int j = 0; j < 8; ++j) {
      float s = rab2[j];
      for (int i = 0; i < 8; ++i) s += a1[i] * raw2[i * 8 + j];
      a2[j] = siluf(s);
    }
#pragma unroll
    for (int j = 0; j < 8; ++j) {
      float s = rab3[j];
      for (int i = 0; i < 8; ++i) s += a2[i] * raw3[i * 8 + j];
      a3[j] = siluf(s);
    }
    float att = rab4[0];
#pragma unroll
    for (int i = 0; i < 8; ++i) att += a3[i] * raw4[i];
    attl[wave][lane] = att;
  }
  __syncthreads();

  const int   mlo    = lane & 15;
  const int   khalf  = (lane >> 4) * 2;
  const int   rowsel = (lane >> 4) * 8;
  const float inv_sq8 = 0.35355339059327373f;
  const float inv_sq3 = 0.57735026918962576f;

  // ---- t = ssp(ele @ fc_w1 / sqrt8) ----
  {
    v8f c = {};
    for (int kk = 0; kk < 2; ++kk) {
      const int k0 = kk * 4 + khalf;
      v2f a; a.x = Aele[wave][mlo * 8 + k0]; a.y = Aele[wave][mlo * 8 + k0 + 1];
      v2f b;
      b.x = (mlo < 8) ? fc_w1[k0 * 8 + mlo] : 0.f;
      b.y = (mlo < 8) ? fc_w1[(k0 + 1) * 8 + mlo] : 0.f;
      c = wmma_f32(a, b, c);
    }
    if (mlo < 8) {
#pragma unroll
      for (int r = 0; r < 8; ++r)
        Tlds[wave][(r + rowsel) * 8 + mlo] = sspf(c[r] * inv_sq8);
    }
  }
  __syncthreads();

  // ---- w GEMM + epilogue, one u-group (16 u's) at a time ----
  for (int ug = 0; ug < 4; ++ug) {
    for (int j = 0; j < 4; ++j) {        // wa/wb/wc/wd slice
      v8f c = {};
      const int col = j * 64 + ug * 16 + mlo;
      for (int kk = 0; kk < 2; ++kk) {
        const int k0 = kk * 4 + khalf;
        v2f a; a.x = Tlds[wave][mlo * 8 + k0]; a.y = Tlds[wave][mlo * 8 + k0 + 1];
        v2f b; b.x = fc_w2[(size_t)k0 * 256 + col];
               b.y = fc_w2[(size_t)(k0 + 1) * 256 + col];
        c = wmma_f32(a, b, c);
      }
#pragma unroll
      for (int r = 0; r < 8; ++r)
        wbuf[wave][(j * 16 + (r + rowsel)) * 16 + mlo] = c[r] * inv_sq8;
    }
    __syncthreads();

    // 16 edges x 16 u's = 256 tasks / 32 lanes
    for (int it = 0; it < 8; ++it) {
      const int task = it * 32 + lane;
      const int e  = task >> 4;
      const int un = task & 15;
      const int u  = ug * 16 + un;
      const int ge = e0 + e;

      const int dst = edge_index[ge];
      const int src = edge_index[E + ge];
      const float att  = attl[wave][e];
      const float sh0  = edge_sh[(size_t)ge * 4 + 0];
      const float sh1x = edge_sh[(size_t)ge * 4 + 1];
      const float sh1y = edge_sh[(size_t)ge * 4 + 2];
      const float sh1z = edge_sh[(size_t)ge * 4 + 3];
      const float wa = wbuf[wave][(0 * 16 + e) * 16 + un];
      const float wb = wbuf[wave][(1 * 16 + e) * 16 + un];
      const float wc = wbuf[wave][(2 * 16 + e) * 16 + un];
      const float wd = wbuf[wave][(3 * 16 + e) * 16 + un];
      const float s0  = h0p[(size_t)src * 64 + u];
      const float s1x = h1p[(size_t)src * 192 + u * 3 + 0];
      const float s1y = h1p[(size_t)src * 192 + u * 3 + 1];
      const float s1z = h1p[(size_t)src * 192 + u * 3 + 2];

      const float p00 = wa * s0 * sh0 * att;
      const float dot = s1x * sh1x + s1y * sh1y + s1z * sh1z;
      const float p11 = wd * dot * inv_sq3 * att;
      atomic_add_f32(&agg0[(size_t)dst * 128 + u], p00);
      atomic_add_f32(&agg0[(size_t)dst * 128 + 64 + u], p11);

      const float wbs = wb * s0 * att;
      atomic_add_f32(&agg1[(size_t)dst * 384 + u * 3 + 0], wbs * sh1x);
      atomic_add_f32(&agg1[(size_t)dst * 384 + u * 3 + 1], wbs * sh1y);
      atomic_add_f32(&agg1[(size_t)dst * 384 + u * 3 + 2], wbs * sh1z);
      const float wcs = wc * sh0 * att;
      atomic_add_f32(&agg1[(size_t)dst * 384 + (64 + u) * 3 + 0], wcs * s1x);
      atomic_add_f32(&agg1[(size_t)dst * 384 + (64 + u) * 3 + 1], wcs * s1y);
      atomic_add_f32(&agg1[(size_t)dst * 384 + (64 + u) * 3 + 2], wcs * s1z);
    }
    __syncthreads();
  }
}

// ---------------------------------------------------------------------------
// Kernel 3: out += agg @ lin2 / sqrt(128).  Block = 128 = 4 waves, one
// 16-node tile; wave 0 -> agg0/lin2_w0, waves 1..3 -> agg1_m/lin2_w1.
// ---------------------------------------------------------------------------
__global__ __launch_bounds__(128) void final_kernel(
    const float* __restrict__ agg0,      // (N,128)
    const float* __restrict__ agg1,      // (N,128,3)
    const float* __restrict__ lin2_w0,   // (128,64)
    const float* __restrict__ lin2_w1,   // (128,64)
    float* __restrict__ out)             // (N,256)
{
  __shared__ float Alds[4][16 * 128];    // 32 KB
  const int row0 = blockIdx.x * 16;
  const int tid  = threadIdx.x;
  const int wave = tid >> 5;
  const int lane = tid & 31;

  for (int i = tid; i < 16 * 128; i += 128) {
    const int r = i >> 7, u = i & 127;
    Alds[0][r * 128 + u] = agg0[(size_t)(row0 + r) * 128 + u];
    Alds[1][r * 128 + u] = agg1[(size_t)(row0 + r) * 384 + u * 3 + 0];
    Alds[2][r * 128 + u] = agg1[(size_t)(row0 + r) * 384 + u * 3 + 1];
    Alds[3][r * 128 + u] = agg1[(size_t)(row0 + r) * 384 + u * 3 + 2];
  }
  __syncthreads();

  const float* A = Alds[wave];
  const float* B = (wave == 0) ? lin2_w0 : lin2_w1;
  const int   mlo    = lane & 15;
  const int   khalf  = (lane >> 4) * 2;
  const int   rowsel = (lane >> 4) * 8;
  const float scale  = 0.08838834764831845f;  // 1/sqrt(128)

  for (int ct = 0; ct < 4; ++ct) {
    v8f c = {};
    const int colbase = ct * 16;
    for (int kk = 0; kk < 32; ++kk) {
      const int k0 = kk * 4 + khalf;
      v2f a; a.x = A[mlo * 128 + k0]; a.y = A[mlo * 128 + k0 + 1];
      v2f b; b.x = B[(size_t)k0 * 64 + colbase + mlo];
             b.y = B[(size_t)(k0 + 1) * 64 + colbase + mlo];
      c = wmma_f32(a, b, c);
    }
#pragma unroll
    for (int r = 0; r < 8; ++r) {
      const int row  = row0 + r + rowsel;
      const int vcol = colbase + mlo;
      const size_t oidx = (wave == 0)
          ? ((size_t)row * 256 + vcol)
          : ((size_t)row * 256 + 64 + vcol * 3 + (wave - 1));
      out[oidx] += c[r] * scale;
    }
  }
}

// ---------------------------------------------------------------------------
extern "C" void kernel_launch(void* const* d_in, const int* in_sizes, int n_in,
                              void* d_out, int out_size, void* d_ws, size_t ws_size,
                              hipStream_t stream) {
  const float* x    = (const float*)d_in[0];
  const float* h    = (const float*)d_in[1];
  const float* ele  = (const float*)d_in[2];
  const float* esh  = (const float*)d_in[3];
  const float* rij  = (const float*)d_in[4];
  const float* l1w0 = (const float*)d_in[5];
  const float* l1w1 = (const float*)d_in[6];
  const float* fcw1 = (const float*)d_in[7];
  const float* fcw2 = (const float*)d_in[8];
  const float* l2w0 = (const float*)d_in[9];
  const float* l2w1 = (const float*)d_in[10];
  const float* scw0 = (const float*)d_in[11];
  const float* scw1 = (const float*)d_in[12];
  const float* raw1 = (const float*)d_in[13];
  const float* rab1 = (const float*)d_in[14];
  const float* raw2 = (const float*)d_in[15];
  const float* rab2 = (const float*)d_in[16];
  const float* raw3 = (const float*)d_in[17];
  const float* rab3 = (const float*)d_in[18];
  const float* raw4 = (const float*)d_in[19];
  const float* rab4 = (const float*)d_in[20];
  const int*   eidx = (const int*)d_in[21];

  const int N = in_sizes[1] / 256;   // 10000
  const int E = in_sizes[2] / 8;     // 160000

  float* ws   = (float*)d_ws;
  float* h0p  = ws;                      // N*64
  float* h1p  = ws + (size_t)N * 64;     // N*192
  float* agg0 = ws + (size_t)N * 256;    // N*128
  float* agg1 = ws + (size_t)N * 384;    // N*384
  float* out  = (float*)d_out;

  const size_t nagg = (size_t)N * 512;
  init_kernel<<<(unsigned)((nagg + 255) / 256), 256, 0, stream>>>(agg0, nagg);

  node_kernel<<<N / 16, 128, 0, stream>>>(x, h, l1w0, l1w1, scw0, scw1,
                                          out, h0p, h1p);

  edge_kernel<<<E / 64, 128, 0, stream>>>(ele, esh, rij, fcw1, fcw2,
                                          raw1, rab1, raw2, rab2,
                                          raw3, rab3, raw4, rab4,
                                          eidx, h0p, h1p, agg0, agg1, E);

  final_kernel<<<N / 16, 128, 0, stream>>>(agg0, agg1, l2w0, l2w1, out);
}